// VectorQuantizer_85615878078791
// MI455X (gfx1250) — compile-verified
//
#include <hip/hip_runtime.h>

typedef float v2f __attribute__((ext_vector_type(2)));
typedef float v8f __attribute__((ext_vector_type(8)));

#define K_CODES        1024
#define DIM            64
#define DPAD           68                // 64 dims + [||e||^2, 0, 0, 0] -> 17 K=4 steps
#define KSTEPS         (DPAD / 4)        // 17
#define HW             1024              // 32*32 spatial positions per image
#define SLICE_ROWS     128               // codebook rows staged in LDS at a time
#define NSLICES        (K_CODES / SLICE_ROWS)
#define ROWS_PER_WAVE  16
#define WAVES_PER_BLK  8
#define ROWS_PER_BLK   (ROWS_PER_WAVE * WAVES_PER_BLK)

__global__ __launch_bounds__(256)
void vq_argmin_gather_kernel(const float* __restrict__ z_e,
                             const float* __restrict__ weight,
                             float* __restrict__ z_q)
{
    // Transformed codebook slice: row n = [-2*e_n (64), ||e_n||^2, 0, 0, 0].
    // 128 * 68 * 4 = 34,816 B; 272B row stride is LDS bank-conflict free.
    __shared__ float lds_w[SLICE_ROWS * DPAD];

    const int tid  = threadIdx.x;
    const int lane = tid & 31;
    const int wave = tid >> 5;
    const int half = lane >> 4;      // 0: lanes 0-15, 1: lanes 16-31
    const int l16  = lane & 15;

    const int row0 = blockIdx.x * ROWS_PER_BLK + wave * ROWS_PER_WAVE;
    const int m    = row0 + l16;     // this lane's A-matrix row (pixel id)
    const int b    = m >> 10;        // image index (m / 1024)
    const int hw   = m & 1023;       // spatial offset (m % 1024)

    // ---- Preload A fragments for V_WMMA_F32_16X16X4_F32 on X' = [x, 1, 0, 0, 0].
    // A layout: M = lane%16, VGPR0/1 hold K = 2*(lane/16) + {0,1}.
    // X[m,d] lives at z_e[b*65536 + d*1024 + hw]  (NCHW).
    v2f a[KSTEPS];
    {
        const float* px = z_e + ((size_t)b << 16) + hw;
        #pragma unroll
        for (int kk = 0; kk < 16; ++kk) {
            const int d0 = 4 * kk + 2 * half;
            a[kk].x = px[(size_t)d0 * HW];
            a[kk].y = px[(size_t)(d0 + 1) * HW];
        }
        // k = 64..67: X'[m,64] = 1 (picks up ||e||^2), rest 0.
        a[16].x = half ? 0.0f : 1.0f;
        a[16].y = 0.0f;
    }

    float minv[8];
    int   mini[8];
    #pragma unroll
    for (int r = 0; r < 8; ++r) { minv[r] = 3.4028235e38f; mini[r] = 0; }

    for (int s = 0; s < NSLICES; ++s) {
        // ---- Cooperative slice load + transform: 2 threads per codebook row.
        {
            const int rrow = tid >> 1;           // 0..127 row within slice
            const int rhd  = tid & 1;            // which 32-float half of the row
            const float4* __restrict__ src = (const float4*)
                (weight + ((size_t)(s * SLICE_ROWS + rrow)) * DIM + rhd * 32);
            float* dstrow = lds_w + rrow * DPAD + rhd * 32;
            float ss = 0.0f;
            #pragma unroll
            for (int i = 0; i < 8; ++i) {
                const float4 v = src[i];         // coalesced global_load_b128
                ss = fmaf(v.x, v.x, fmaf(v.y, v.y, fmaf(v.z, v.z, fmaf(v.w, v.w, ss))));
                float4 o;
                o.x = -2.0f * v.x; o.y = -2.0f * v.y;
                o.z = -2.0f * v.z; o.w = -2.0f * v.w;
                ((float4*)dstrow)[i] = o;        // ds_store_b128
            }
            // Combine the two half-row sums (adjacent lanes, same wave).
            ss += __shfl_xor(ss, 1, 32);
            if (rhd == 0) {
                dstrow[64] = ss;                 // ||e||^2 at column 64
            } else {
                dstrow[33] = 0.0f;               // columns 65..67 = 0 (pad)
                dstrow[34] = 0.0f;
                dstrow[35] = 0.0f;
            }
        }
        __syncthreads();

        #pragma unroll 1
        for (int j = 0; j < SLICE_ROWS / 16; ++j) {
            // B fragment mirrors A with M->N: N = lane%16, K = 2*(lane/16)+{0,1}.
            const int nloc = j * 16 + l16;
            const v2f* __restrict__ brow = (const v2f*)(lds_w + nloc * DPAD);

            v8f acc = {0.f, 0.f, 0.f, 0.f, 0.f, 0.f, 0.f, 0.f};
            #pragma unroll
            for (int kk = 0; kk < KSTEPS; ++kk) {
                const v2f bb = brow[2 * kk + half];    // ds_load_b64
                acc = __builtin_amdgcn_wmma_f32_16x16x4_f32(
                        false, a[kk], false, bb, (short)0, acc, false, false);
            }
            // acc[r] IS the distance (minus the constant ||x||^2 term).
            // C/D layout: lane holds column n, VGPR r holds row m = r + 8*half.
            const int nidx = s * SLICE_ROWS + nloc;
            #pragma unroll
            for (int r = 0; r < 8; ++r) {
                const float d = acc[r];
                if (d < minv[r]) { minv[r] = d; mini[r] = nidx; }  // first-win tiebreak
            }
        }
        __syncthreads();
    }

    // ---- Cross-lane argmin within each 16-lane half (columns of the C tile).
    #pragma unroll
    for (int r = 0; r < 8; ++r) {
        #pragma unroll
        for (int off = 1; off < 16; off <<= 1) {
            const float ov = __shfl_xor(minv[r], off, 32);
            const int   oi = __shfl_xor(mini[r], off, 32);
            if (ov < minv[r] || (ov == minv[r] && oi < mini[r])) {
                minv[r] = ov; mini[r] = oi;
            }
        }
    }
    // lanes 0-15 now all hold best idx for pixels row0+r;
    // lanes 16-31 hold best idx for pixels row0+8+r (r = VGPR index 0..7).

    // ---- Broadcast so lane L (both halves) learns index of pixel p = L%16.
    int myIdx = 0;
    #pragma unroll
    for (int r = 0; r < 8; ++r) {
        const int id0 = __shfl(mini[r], 0,  32);   // pixels 0..7
        const int id1 = __shfl(mini[r], 16, 32);   // pixels 8..15
        if (l16 == r)     myIdx = id0;
        if (l16 == r + 8) myIdx = id1;
    }

    // ---- Gather codebook row and write z_q in NCHW.
    // Lane handles pixel p = l16 (16 consecutive w positions per half-wave),
    // channels c = 2*cc + half  ->  contiguous 64B stores per half-wave.
    {
        const float* __restrict__ wrow = weight + (size_t)myIdx * DIM;
        float* pout = z_q + (((size_t)b * DIM) * HW) + (size_t)(row0 & 1023) + l16;
        #pragma unroll
        for (int cc = 0; cc < 32; ++cc) {
            const int c = 2 * cc + half;
            pout[(size_t)c * HW] = wrow[c];
        }
    }
}

extern "C" void kernel_launch(void* const* d_in, const int* in_sizes, int n_in,
                              void* d_out, int out_size, void* d_ws, size_t ws_size,
                              hipStream_t stream) {
    const float* z_e    = (const float*)d_in[0];   // [64, 64, 32, 32] fp32
    const float* weight = (const float*)d_in[1];   // [1024, 64] fp32
    float*       z_q    = (float*)d_out;           // [64, 64, 32, 32] fp32

    const int total_rows = 64 * 32 * 32;               // 65536 pixel vectors
    const int grid       = total_rows / ROWS_PER_BLK;  // 512 blocks

    vq_argmin_gather_kernel<<<grid, 256, 0, stream>>>(z_e, weight, z_q);
}